// GetMaskedAnchorBoxes_46574625357913
// MI455X (gfx1250) — compile-verified
//
#include <hip/hip_runtime.h>
#include <stdint.h>

#define K_NA 3
#define K_NM 32
#define K_NC 80
#define K_CH 351          // NA*(5+NC) + NA*NM
#define K_NHW 6400
#define K_W 80
#define K_BLK 256
#define K_CPB 25          // NHW / BLK
#define K_THRESH 0.05f
#define K_STRIDE 8.0f
#define K_ROW (7 + K_NM)  // 39

__device__ __forceinline__ float sigmoidf_(float x) { return 1.0f / (1.0f + expf(-x)); }

// ---------------------------------------------------------------------------
// Pass 1: per-cell score = softmax_max(cls) * sigmoid(obj), plus argmax class
// and a per-block keep count. The 80 class-logit channels are streamed
// global -> LDS with the CDNA5 async DMA path (ASYNCcnt), double buffered
// per wave (16 channels per chunk, 5 chunks).
// ---------------------------------------------------------------------------
__global__ void __launch_bounds__(K_BLK)
yolo_score_pass(const float* __restrict__ net, float* __restrict__ score,
                float* __restrict__ clsf, unsigned* __restrict__ blkCnt)
{
    __shared__ float stage[K_BLK / 32][2][16][32];   // [wave][buf][chan][lane]
    __shared__ unsigned red[K_BLK];

    const int t     = threadIdx.x;
    const int blk   = blockIdx.x;
    const int ba    = blk / K_CPB;           // b*NA + a
    const int chunk = blk - ba * K_CPB;
    const int b     = ba / K_NA;
    const int a     = ba - b * K_NA;
    const int p     = chunk * K_BLK + t;
    const int wave  = t >> 5, lane = t & 31;

    const float* boxBase = net + ((size_t)b * K_CH + (size_t)a * (5 + K_NC)) * K_NHW;
    const float* cls0    = boxBase + (size_t)5 * K_NHW + p;   // per-thread ptr, class 0

    auto issue16 = [&](int k, int buf) {
#pragma unroll
        for (int c = 0; c < 16; ++c) {
            unsigned long long g =
                (unsigned long long)(uintptr_t)(cls0 + (size_t)(k * 16 + c) * K_NHW);
            unsigned lofs = (unsigned)(uintptr_t)&stage[wave][buf][c][lane];
            asm volatile("global_load_async_to_lds_b32 %0, %1, off"
                         :: "v"(lofs), "v"(g) : "memory");
        }
    };

    issue16(0, 0);

    float m = -__builtin_inff();
    float s = 0.0f;
    int   idx = 0;
#pragma unroll
    for (int k = 0; k < 5; ++k) {
        if (k < 4) {
            issue16(k + 1, (k + 1) & 1);                         // prefetch next chunk
            asm volatile("s_wait_asynccnt 16" ::: "memory");     // current chunk landed
        } else {
            asm volatile("s_wait_asynccnt 0" ::: "memory");
        }
        const int buf = k & 1;
#pragma unroll
        for (int c = 0; c < 16; ++c) {
            float l = stage[wave][buf][c][lane];
            if (l > m) { s = s * expf(m - l) + 1.0f; m = l; idx = k * 16 + c; }
            else       { s += expf(l - m); }
        }
    }

    const float obj = sigmoidf_(boxBase[(size_t)4 * K_NHW + p]);
    const float sc  = obj / s;                 // softmax_max = 1/s

    const size_t cell = (size_t)blk * K_BLK + t;
    score[cell] = sc;
    clsf[cell]  = (float)idx;

    unsigned keep = sc > K_THRESH ? 1u : 0u;
    red[t] = keep;
    __syncthreads();
#pragma unroll
    for (int o = K_BLK / 2; o > 0; o >>= 1) {
        if (t < o) red[t] += red[t + o];
        __syncthreads();
    }
    if (t == 0) blkCnt[blk] = red[0];
}

// ---------------------------------------------------------------------------
// Pass 2: exclusive scan over the per-block keep counts (single block).
// ---------------------------------------------------------------------------
__global__ void __launch_bounds__(256)
yolo_scan_blocks(const unsigned* __restrict__ cnt, unsigned* __restrict__ off, int n)
{
    __shared__ unsigned ssum[256];
    const int t = threadIdx.x;
    const int PER = (n + 255) >> 8;          // 5 for n=1200
    unsigned local[8];
    unsigned run = 0;
#pragma unroll
    for (int i = 0; i < 8; ++i) {
        if (i < PER) {
            int idx = t * PER + i;
            unsigned v = (idx < n) ? cnt[idx] : 0u;
            local[i] = run;
            run += v;
        }
    }
    ssum[t] = run;
    __syncthreads();
    for (int o = 1; o < 256; o <<= 1) {
        unsigned v = (t >= o) ? ssum[t - o] : 0u;
        __syncthreads();
        ssum[t] += v;
        __syncthreads();
    }
    unsigned base = (t == 0) ? 0u : ssum[t - 1];
#pragma unroll
    for (int i = 0; i < 8; ++i) {
        if (i < PER) {
            int idx = t * PER + i;
            if (idx < n) off[idx] = base + local[i];
        }
    }
}

// ---------------------------------------------------------------------------
// Pass 3: ordered compaction. Block-local exclusive scan of keep flags gives
// the rank; kept cells decode coords + gather the 32 mask coefficients (L2
// hits: whole input < 192MB L2) into a dense LDS row buffer; the compacted
// region is then drained to global memory COALESCED via the CDNA5 async
// LDS->global DMA (global_store_async_from_lds_b32, ASYNCcnt).
// ---------------------------------------------------------------------------
__global__ void __launch_bounds__(K_BLK)
yolo_emit_pass(const float* __restrict__ net, const float* __restrict__ anchors,
               const float* __restrict__ score, const float* __restrict__ clsf,
               const unsigned* __restrict__ blkOff, float* __restrict__ dst)
{
    __shared__ unsigned sc[K_BLK];
    __shared__ float rows[K_BLK * K_ROW];   // dense compacted rows, ~39KB
    const int t     = threadIdx.x;
    const int blk   = blockIdx.x;
    const int ba    = blk / K_CPB;
    const int chunk = blk - ba * K_CPB;
    const int b     = ba / K_NA;
    const int a     = ba - b * K_NA;
    const int p     = chunk * K_BLK + t;

    const size_t cell = (size_t)blk * K_BLK + t;
    const float scv = score[cell];
    const unsigned keep = scv > K_THRESH ? 1u : 0u;

    const float* maskBase =
        net + ((size_t)b * K_CH + (size_t)(K_NA * (5 + K_NC)) + (size_t)a * K_NM) * K_NHW + p;
    if (keep) {  // warm the scattered mask gathers while the scan runs
        __builtin_prefetch(maskBase, 0, 0);
        __builtin_prefetch(maskBase + (size_t)16 * K_NHW, 0, 0);
    }

    sc[t] = keep;
    __syncthreads();
    for (int o = 1; o < K_BLK; o <<= 1) {
        unsigned v = (t >= o) ? sc[t - o] : 0u;
        __syncthreads();
        sc[t] += v;
        __syncthreads();
    }
    const unsigned rank = sc[t] - keep;

    if (keep) {
        float* r = rows + rank * K_ROW;

        const float* boxBase =
            net + ((size_t)b * K_CH + (size_t)a * (5 + K_NC)) * K_NHW + p;
        const float tx = boxBase[0];
        const float ty = boxBase[(size_t)1 * K_NHW];
        const float tw = boxBase[(size_t)2 * K_NHW];
        const float th = boxBase[(size_t)3 * K_NHW];

        float amax = anchors[0];
#pragma unroll
        for (int i = 1; i < 2 * K_NA; ++i) amax = fmaxf(amax, anchors[i]);
        const float mv = floorf(logf(1e35f / amax / K_STRIDE));

        const float aw = anchors[a * 2 + 0];
        const float ah = anchors[a * 2 + 1];
        const float lx = (float)(p % K_W);
        const float ly = (float)(p / K_W);

        r[0] = (float)b;
        r[1] = (sigmoidf_(tx) + lx) * K_STRIDE;
        r[2] = (sigmoidf_(ty) + ly) * K_STRIDE;
        r[3] = expf(fminf(tw, mv)) * aw * K_STRIDE;
        r[4] = expf(fminf(th, mv)) * ah * K_STRIDE;
#pragma unroll
        for (int mi = 0; mi < K_NM; ++mi) r[5 + mi] = maskBase[(size_t)mi * K_NHW];
        r[5 + K_NM] = scv;
        r[6 + K_NM] = clsf[cell];
    }
    __syncthreads();   // LDS rows complete before the async engine reads them

    const unsigned n = sc[K_BLK - 1] * K_ROW;          // dwords to emit
    float* outBase = dst + (size_t)blkOff[blk] * K_ROW;
    for (unsigned i = t; i < n; i += K_BLK) {          // coalesced LDS -> global DMA
        unsigned long long g = (unsigned long long)(uintptr_t)(outBase + i);
        unsigned l = (unsigned)(uintptr_t)&rows[i];
        asm volatile("global_store_async_from_lds_b32 %0, %1, off"
                     :: "v"(g), "v"(l) : "memory");
    }
    asm volatile("s_wait_asynccnt 0" ::: "memory");    // (s_endpgm would also wait)
}

// ---------------------------------------------------------------------------
extern "C" void kernel_launch(void* const* d_in, const int* in_sizes, int n_in,
                              void* d_out, int out_size, void* d_ws, size_t ws_size,
                              hipStream_t stream) {
    const float* net     = (const float*)d_in[0];
    const float* anchors = (const float*)d_in[1];

    const int nB      = in_sizes[0] / (K_CH * K_NHW);     // 16
    const int nBlocks = nB * K_NA * K_CPB;                // 1200
    const size_t cells = (size_t)nBlocks * K_BLK;         // 307200

    float*    score  = (float*)d_ws;
    float*    clsf   = score + cells;
    unsigned* blkCnt = (unsigned*)(clsf + cells);
    unsigned* blkOff = blkCnt + nBlocks;

    yolo_score_pass<<<nBlocks, K_BLK, 0, stream>>>(net, score, clsf, blkCnt);
    yolo_scan_blocks<<<1, 256, 0, stream>>>(blkCnt, blkOff, nBlocks);
    yolo_emit_pass<<<nBlocks, K_BLK, 0, stream>>>(net, anchors, score, clsf, blkOff,
                                                  (float*)d_out);
}